// GAT_34368328302697
// MI455X (gfx1250) — compile-verified
//
#include <hip/hip_runtime.h>

typedef __attribute__((ext_vector_type(2))) float v2f;
typedef __attribute__((ext_vector_type(8))) float v8f;

#define NEG_SLOPE 0.2f

// -------------------------------------------------------------------------
// GEMM: xl/xr [N x 32] = X [N x K] @ Wl/Wr [K x 32]  via V_WMMA_F32_16X16X4_F32
// One wave (32 lanes) computes a 16-row x 32-col tile for BOTH Wl and Wr,
// sharing the A-matrix fragment loads. 4 waves per 128-thread block.
//
// A 16x4 f32 fragment layout (per ISA): lanes 0-15 -> M=lane, VGPR0=K0,VGPR1=K1;
// lanes 16-31 -> M=lane-16, VGPR0=K2, VGPR1=K3.
// B 4x16 layout mirrors A with N instead of M.
// C/D 16x16 f32: VGPR r: lanes 0-15 -> M=r,N=lane; lanes 16-31 -> M=r+8,N=lane-16.
// -------------------------------------------------------------------------
template <int K>
__global__ void gatv2_gemm_lr(const float* __restrict__ X,
                              const float* __restrict__ Wl,
                              const float* __restrict__ Wr,
                              float* __restrict__ xl,
                              float* __restrict__ xr,
                              int ntiles) {
  const int wave = threadIdx.x >> 5;
  const int tile = blockIdx.x * (blockDim.x >> 5) + wave;
  if (tile >= ntiles) return;  // uniform per wave: EXEC stays all-ones for WMMA
  const int lane = threadIdx.x & 31;
  const int half = lane >> 4;   // 0: lanes 0-15, 1: lanes 16-31
  const int lid = lane & 15;

  const float* xrow = X + (size_t)(tile * 16 + lid) * K;

  v8f cl0 = {}, cl1 = {}, cr0 = {}, cr1 = {};
#pragma unroll
  for (int k = 0; k < K; k += 4) {
    const int kk = k + 2 * half;
    v2f a;
    a.x = xrow[kk];
    a.y = xrow[kk + 1];
    v2f bl0, bl1, br0, br1;
    bl0.x = Wl[kk * 32 + lid];       bl0.y = Wl[(kk + 1) * 32 + lid];
    bl1.x = Wl[kk * 32 + 16 + lid];  bl1.y = Wl[(kk + 1) * 32 + 16 + lid];
    br0.x = Wr[kk * 32 + lid];       br0.y = Wr[(kk + 1) * 32 + lid];
    br1.x = Wr[kk * 32 + 16 + lid];  br1.y = Wr[(kk + 1) * 32 + 16 + lid];
    // 8 args: (neg_a, A, neg_b, B, c_mod, C, reuse_a, reuse_b)
    cl0 = __builtin_amdgcn_wmma_f32_16x16x4_f32(false, a, false, bl0, (short)0, cl0, false, false);
    cl1 = __builtin_amdgcn_wmma_f32_16x16x4_f32(false, a, false, bl1, (short)0, cl1, false, false);
    cr0 = __builtin_amdgcn_wmma_f32_16x16x4_f32(false, a, false, br0, (short)0, cr0, false, false);
    cr1 = __builtin_amdgcn_wmma_f32_16x16x4_f32(false, a, false, br1, (short)0, cr1, false, false);
  }

#pragma unroll
  for (int r = 0; r < 8; ++r) {
    const int m = r + 8 * half;
    const size_t o = (size_t)(tile * 16 + m) * 32;
    xl[o + lid] = cl0[r];
    xl[o + 16 + lid] = cl1[r];
    xr[o + lid] = cr0[r];
    xr[o + 16 + lid] = cr1[r];
  }
}

// -------------------------------------------------------------------------
// Zero acc[N*32], denom[N]; maxenc[N] = 0 (0 == encoding below -inf's code,
// safe because every node has a self-loop so every max gets a real score).
// -------------------------------------------------------------------------
__global__ void gatv2_init(float* __restrict__ acc, float* __restrict__ denom,
                           unsigned int* __restrict__ maxenc, int nnodes) {
  const int i = blockIdx.x * blockDim.x + threadIdx.x;
  if (i < nnodes * 32) acc[i] = 0.0f;
  if (i < nnodes) {
    denom[i] = 0.0f;
    maxenc[i] = 0u;
  }
}

__device__ __forceinline__ unsigned int enc_f32(float f) {
  unsigned int u = __float_as_uint(f);
  return (u & 0x80000000u) ? ~u : (u | 0x80000000u);
}
__device__ __forceinline__ float dec_f32(unsigned int u) {
  return __uint_as_float((u & 0x80000000u) ? (u & 0x7fffffffu) : ~u);
}

// -------------------------------------------------------------------------
// Pass 1 over edges: score[e] = att . leakyrelu(xl[src] + xr[dst]);
// segment max via monotonic-uint atomicMax. 8 lanes per edge, float4/lane,
// so each gather is one contiguous 128B line (L2-resident: xl/xr = 12.8MB).
// -------------------------------------------------------------------------
__global__ void gatv2_score(const long long* __restrict__ ei,  // [2, E]
                            const float* __restrict__ xl,
                            const float* __restrict__ xr,
                            const float* __restrict__ att,
                            float* __restrict__ score,
                            unsigned int* __restrict__ maxenc, int E,
                            int nnodes) {
  const long long tid = (long long)blockIdx.x * blockDim.x + threadIdx.x;
  const int e = (int)(tid >> 3);
  const int g = (int)(tid & 7);
  const int etot = E + nnodes;
  if (e >= etot) return;
  int s, d;
  if (e < E) {
    s = (int)ei[e];
    d = (int)ei[(size_t)E + e];
  } else {
    s = d = e - E;  // self loop
  }
  const float* ps = xl + (size_t)s * 32 + g * 4;
  const float* pd = xr + (size_t)d * 32 + g * 4;
  const float* pa = att + g * 4;
  float p = 0.0f;
#pragma unroll
  for (int j = 0; j < 4; ++j) {
    float v = ps[j] + pd[j];
    v = (v > 0.0f) ? v : v * NEG_SLOPE;
    p = fmaf(pa[j], v, p);
  }
  p += __shfl_xor(p, 4, 8);
  p += __shfl_xor(p, 2, 8);
  p += __shfl_xor(p, 1, 8);
  if (g == 0) {
    score[e] = p;
    atomicMax(&maxenc[d], enc_f32(p));
  }
}

// -------------------------------------------------------------------------
// Pass 2 over edges: a = exp(score - max[dst]);
// denom[dst] += a;  acc[dst][:] += a * xl[src][:]   (unnormalized numerator)
// -------------------------------------------------------------------------
__global__ void gatv2_accum(const long long* __restrict__ ei,
                            const float* __restrict__ xl,
                            const float* __restrict__ score,
                            const unsigned int* __restrict__ maxenc,
                            float* __restrict__ denom, float* __restrict__ acc,
                            int E, int nnodes) {
  const long long tid = (long long)blockIdx.x * blockDim.x + threadIdx.x;
  const int e = (int)(tid >> 3);
  const int g = (int)(tid & 7);
  const int etot = E + nnodes;
  if (e >= etot) return;
  int s, d;
  if (e < E) {
    s = (int)ei[e];
    d = (int)ei[(size_t)E + e];
  } else {
    s = d = e - E;
  }
  const float a = expf(score[e] - dec_f32(maxenc[d]));
  if (g == 0) atomicAdd(&denom[d], a);
  const float* ps = xl + (size_t)s * 32 + g * 4;
  float* pacc = acc + (size_t)d * 32 + g * 4;
#pragma unroll
  for (int j = 0; j < 4; ++j) atomicAdd(&pacc[j], a * ps[j]);
}

// -------------------------------------------------------------------------
// Per-node finalize: out = acc/denom + bias (optional ReLU)
// -------------------------------------------------------------------------
__global__ void gatv2_finish(const float* __restrict__ acc,
                             const float* __restrict__ denom,
                             const float* __restrict__ bias,
                             float* __restrict__ out, int nnodes, int do_relu) {
  const int i = blockIdx.x * blockDim.x + threadIdx.x;
  if (i >= nnodes * 32) return;
  const int node = i >> 5;
  const int col = i & 31;
  float v = acc[i] / denom[node] + bias[col];
  if (do_relu) v = fmaxf(v, 0.0f);
  out[i] = v;
}

// -------------------------------------------------------------------------
// inputs (setup_inputs order):
// 0:x [N,128] f32   1:positive_edge_index [2,E] i64
// 2:Wl1[128,32] 3:Wr1[128,32] 4:att1[32] 5:b1[32]
// 6:Wl2[32,32]  7:Wr2[32,32]  8:att2[32] 9:b2[32]
// output: [N,32] f32
// -------------------------------------------------------------------------
extern "C" void kernel_launch(void* const* d_in, const int* in_sizes, int n_in,
                              void* d_out, int out_size, void* d_ws,
                              size_t ws_size, hipStream_t stream) {
  const float* x = (const float*)d_in[0];
  const long long* ei = (const long long*)d_in[1];
  const float* Wl1 = (const float*)d_in[2];
  const float* Wr1 = (const float*)d_in[3];
  const float* att1 = (const float*)d_in[4];
  const float* b1 = (const float*)d_in[5];
  const float* Wl2 = (const float*)d_in[6];
  const float* Wr2 = (const float*)d_in[7];
  const float* att2 = (const float*)d_in[8];
  const float* b2 = (const float*)d_in[9];

  const int N = in_sizes[0] / 128;  // 100000 (divisible by 16)
  const int E = in_sizes[1] / 2;    // 1600000

  // workspace carve-up (~60 MB)
  float* ws = (float*)d_ws;
  size_t o = 0;
  float* xl = ws + o;    o += (size_t)N * 32;
  float* xr = ws + o;    o += (size_t)N * 32;
  float* acc = ws + o;   o += (size_t)N * 32;
  float* h = ws + o;     o += (size_t)N * 32;
  float* score = ws + o; o += (size_t)(E + N);
  float* denom = ws + o; o += (size_t)N;
  unsigned int* maxenc = (unsigned int*)(ws + o);

  const int ntiles = N / 16;
  const int ggrid = (ntiles + 3) / 4;  // 4 waves / 128-thread block
  const int ngrid = (N * 32 + 255) / 256;
  const long long ethreads = (long long)(E + N) * 8;
  const int egrid = (int)((ethreads + 255) / 256);

  // ---- layer 1: DIN=128 -> DH=32, ReLU ----
  gatv2_gemm_lr<128><<<ggrid, 128, 0, stream>>>(x, Wl1, Wr1, xl, xr, ntiles);
  gatv2_init<<<ngrid, 256, 0, stream>>>(acc, denom, maxenc, N);
  gatv2_score<<<egrid, 256, 0, stream>>>(ei, xl, xr, att1, score, maxenc, E, N);
  gatv2_accum<<<egrid, 256, 0, stream>>>(ei, xl, score, maxenc, denom, acc, E, N);
  gatv2_finish<<<ngrid, 256, 0, stream>>>(acc, denom, b1, h, N, 1);

  // ---- layer 2: DH=32 -> DOUT=32 ----
  gatv2_gemm_lr<32><<<ggrid, 128, 0, stream>>>(h, Wl2, Wr2, xl, xr, ntiles);
  gatv2_init<<<ngrid, 256, 0, stream>>>(acc, denom, maxenc, N);
  gatv2_score<<<egrid, 256, 0, stream>>>(ei, xl, xr, att2, score, maxenc, E, N);
  gatv2_accum<<<egrid, 256, 0, stream>>>(ei, xl, score, maxenc, denom, acc, E, N);
  gatv2_finish<<<ngrid, 256, 0, stream>>>(acc, denom, b2, (float*)d_out, N, 0);
}